// S4DKernel_12773232738523
// MI455X (gfx1250) — compile-verified
//
#include <hip/hip_runtime.h>
#include <hip/hip_bf16.h>

// ---------------------------------------------------------------------------
// S4D diagonal SSM on MI455X (gfx1250), chunked-convolution formulation.
//
//   h_t = Abar * h_{t-1} + Bbar * x_t        (diagonal, per (d,n))
//   y_t = sum_n C[d,n] * h_t[d,n]
//
// Chunk size T=16, NC=128 chunks, COLS = BATCH*NC = 256 columns.
// Per channel d:
//   Y (16 x 256) = [ Toeplitz(k_d) | G_d ] (16 x 80)  @  [ X_d ; H0_d ] (80 x 256)
// computed with v_wmma_f32_16x16x4_f32. Only the h-carry scan stays serial.
//
// CDNA5 data movement: all LDS staging uses GLOBAL_LOAD_ASYNC_TO_LDS_B32 and
// the final output store uses GLOBAL_STORE_ASYNC_FROM_LDS_B32 (ASYNCcnt +
// s_wait_asynccnt), keeping VALU/VGPRs free around the WMMA and scan loops.
//
// Kernels:
//   K1 s4d_params    : Abar/Bbar, k taps, M = [Toeplitz(k)|G]   (512 x 32)
//   K2 s4d_scan_h0   : full h scan, record h at chunk starts     (256 x 256)
//   K3 s4d_wmma_ymat : per-channel WMMA matmul -> Yws [d][b*L+t] (512 x 256)
//   K4 s4d_transpose : Yws (D x BT) -> out (BT x D), coalesced   (tiled)
//
// Workspace (floats): Abar 32768 | Bbar 32768 | M 655360 | H0 8388608 |
//                     Yws 2097152  => ~44.8 MB total.
// ---------------------------------------------------------------------------

#define D_MODEL 512
#define D_STATE 64
#define BATCH   2
#define LENGTH  2048
#define T       16
#define NC      (LENGTH / T)        // 128
#define COLS    (BATCH * NC)        // 256
#define KDIM    (T + D_STATE)       // 80

typedef float v2f __attribute__((ext_vector_type(2)));
typedef float v8f __attribute__((ext_vector_type(8)));

// --- CDNA5 async global<->LDS helpers --------------------------------------
// Generic LDS pointer = {shared aperture, offset32}; low 32 bits are the LDS
// byte address the VGLOBAL async ops expect in their LDS-address VGPR.
__device__ __forceinline__ void async_g2l_b32(const void* gptr, void* ldsptr) {
  unsigned loff = (unsigned)(unsigned long long)(uintptr_t)ldsptr;
  asm volatile("global_load_async_to_lds_b32 %0, %1, off"
               :: "v"(loff), "v"(gptr)
               : "memory");
}
__device__ __forceinline__ void async_l2g_b32(void* gptr, const void* ldsptr) {
  unsigned loff = (unsigned)(unsigned long long)(uintptr_t)ldsptr;
  asm volatile("global_store_async_from_lds_b32 %0, %1, off"
               :: "v"(gptr), "v"(loff)
               : "memory");
}
__device__ __forceinline__ void wait_async0() {
#if __has_builtin(__builtin_amdgcn_s_wait_asynccnt)
  __builtin_amdgcn_s_wait_asynccnt(0);
#else
  asm volatile("s_wait_asynccnt 0x0" ::: "memory");
#endif
}

// ---------------------------------------------------------------------------
// K1: per-channel parameters + combined matrix M = [Toeplitz(k) | G]
//     k[t]   = sum_n C*Bbar*Abar^t          (16 taps)
//     G[t,n] = C_n * Abar_n^{t+1}
// One wave32 per channel; lane l owns states l and l+32.
// ---------------------------------------------------------------------------
__global__ __launch_bounds__(32) void s4d_params(
    const float* __restrict__ A_log, const float* __restrict__ Bmat,
    const float* __restrict__ Cmat,  const float* __restrict__ log_dt,
    float* __restrict__ Abar, float* __restrict__ Bbar,
    float* __restrict__ Mws) {
  const int d = blockIdx.x;
  const int l = threadIdx.x;
  const int n0 = l, n1 = l + 32;

  const float dt = expf(log_dt[d]);
  const float A0 = expf(-dt * expf(A_log[n0]));
  const float A1 = expf(-dt * expf(A_log[n1]));
  const float B0 = dt * Bmat[d * D_STATE + n0];
  const float B1 = dt * Bmat[d * D_STATE + n1];
  const float C0 = Cmat[d * D_STATE + n0];
  const float C1 = Cmat[d * D_STATE + n1];

  Abar[d * D_STATE + n0] = A0;  Abar[d * D_STATE + n1] = A1;
  Bbar[d * D_STATE + n0] = B0;  Bbar[d * D_STATE + n1] = B1;

  const float CB0 = C0 * B0, CB1 = C1 * B1;
  __shared__ float kt[T];

  float p0 = 1.f, p1 = 1.f;               // p = Abar^t
  for (int t = 0; t < T; ++t) {
    float s = CB0 * p0 + CB1 * p1;        // partial of k[t]
    for (int off = 16; off; off >>= 1)    // wave32 butterfly reduce
      s += __shfl_xor(s, off, 32);
    if (l == 0) kt[t] = s;
    // G columns 16..79 of M:  C * Abar^{t+1}
    Mws[(d * T + t) * KDIM + T + n0] = C0 * A0 * p0;
    Mws[(d * T + t) * KDIM + T + n1] = C1 * A1 * p1;
    p0 *= A0;  p1 *= A1;
  }
  __syncthreads();
  // Toeplitz columns 0..15:  M[i][j] = (i>=j) ? k[i-j] : 0
  for (int q = 0; q < 8; ++q) {
    const int e = l * 8 + q;
    const int i = e >> 4, j = e & 15;
    Mws[(d * T + i) * KDIM + j] = (i >= j) ? kt[i - j] : 0.f;
  }
}

// ---------------------------------------------------------------------------
// K2: serial h scan (1 FMA/state/step), recording h at every chunk start.
// Block = 4 channels x 64 states (8 wave32s); x staged in LDS (64x reuse)
// via async-to-LDS; next chunk prefetched into L2.
// H0 layout: [col = b*NC+c][d][n]  -> fully coalesced 1KB stores per block.
// ---------------------------------------------------------------------------
__global__ __launch_bounds__(256) void s4d_scan_h0(
    const float* __restrict__ x, const float* __restrict__ Abar,
    const float* __restrict__ Bbar, float* __restrict__ H0) {
  const int b  = blockIdx.x >> 7;          // 0..1
  const int d0 = (blockIdx.x & 127) * 4;   // channel group
  const int tid = threadIdx.x;
  const int n = tid & 63, dl = tid >> 6;
  const int d = d0 + dl;

  const float Ab = Abar[d * D_STATE + n];
  const float Bb = Bbar[d * D_STATE + n];

  __shared__ float xs[4 * T];              // 4 channels x 16 steps
  float h = 0.f;

  for (int c = 0; c < NC; ++c) {
    // h at the *start* of chunk c (h0 for chunk c)
    H0[(b * NC + c) * (D_MODEL * D_STATE) + d * D_STATE + n] = h;

    if (tid < 64) {
      const int tl = tid >> 2, dd = tid & 3;     // 16B-contiguous loads
      async_g2l_b32(&x[(b * LENGTH + c * T + tl) * D_MODEL + d0 + dd],
                    &xs[dd * T + tl]);
      if (c + 1 < NC)                            // -> global_prefetch_b8
        __builtin_prefetch(&x[(b * LENGTH + (c + 1) * T + tl) * D_MODEL + d0 + dd], 0, 1);
    }
    wait_async0();
    __syncthreads();
#pragma unroll
    for (int i = 0; i < T; ++i)
      h = fmaf(Ab, h, Bb * xs[dl * T + i]);      // LDS broadcast read
    __syncthreads();
  }
}

// ---------------------------------------------------------------------------
// K3: per-channel WMMA matmul  Y(16x256) = M(16x80) @ Z(80x256)
// Block = 1 channel, 8 wave32s; batch b handled as column half (128 cols).
// Wave w computes N-tile w of its half; K = 80 -> 20 wmma_f32_16x16x4 steps.
// M staged with row stride 81 (81*idx mod 64 injective -> no bank conflicts).
// All staging goes through GLOBAL_LOAD_ASYNC_TO_LDS_B32.
// ---------------------------------------------------------------------------
__global__ __launch_bounds__(256) void s4d_wmma_ymat(
    const float* __restrict__ x, const float* __restrict__ Mws,
    const float* __restrict__ H0, float* __restrict__ Yws) {
  const int d = blockIdx.x;
  __shared__ float Msh[16 * 81];          // 16 x 80, padded
  __shared__ float Zsh[KDIM * 128];       // 80 x 128 column-half

  const int tid  = threadIdx.x;
  const int lane = tid & 31, wave = tid >> 5;
  const int idx  = lane & 15, h2 = lane >> 4;

  for (int e = tid; e < 16 * KDIM; e += 256) {
    const int i = e / KDIM, j = e % KDIM;
    async_g2l_b32(&Mws[(d * T + i) * KDIM + j], &Msh[i * 81 + j]);
  }

  for (int half = 0; half < BATCH; ++half) {   // half == batch index
    __syncthreads();
    // Stage Z: rows 0..15 = x within chunk, rows 16..79 = h0 states
    for (int e = tid; e < KDIM * 128; e += 256) {
      const int j = e >> 7, cl = e & 127;      // row, chunk-in-half
      const float* src;
      if (j < T) {
        src = &x[(half * LENGTH + cl * T + j) * D_MODEL + d];
      } else {
        src = &H0[(half * NC + cl) * (D_MODEL * D_STATE) + d * D_STATE + (j - T)];
      }
      async_g2l_b32(src, &Zsh[j * 128 + cl]);
    }
    wait_async0();                             // covers M stage on half==0 too
    __syncthreads();

    const int cl0 = wave * 16;                 // this wave's 16 columns
    v8f acc = {};
#pragma unroll
    for (int k = 0; k < KDIM / 4; ++k) {       // 20 K-steps of 4
      const int kb = 4 * k + 2 * h2;
      v2f a, bf;
      // A 16x4 layout: lanes 0-15 hold K=kb+0/kb+1, lanes 16-31 K=kb+2/kb+3
      a.x  = Msh[idx * 81 + kb];
      a.y  = Msh[idx * 81 + kb + 1];
      // B 4x16 layout mirrored: row = K, striped across lanes by N
      bf.x = Zsh[kb * 128 + cl0 + idx];
      bf.y = Zsh[(kb + 1) * 128 + cl0 + idx];
      acc = __builtin_amdgcn_wmma_f32_16x16x4_f32(
          /*neg_a=*/false, a, /*neg_b=*/false, bf,
          /*c_mod=*/(short)0, acc, /*reuse_a=*/false, /*reuse_b=*/false);
    }

    // D layout: VGPR r -> row i = r + 8*h2, col N = idx. 8 consecutive
    // time steps per lane -> contiguous 32B stores into Yws[d][b*L + t].
    const int c = cl0 + idx;
#pragma unroll
    for (int r = 0; r < 8; ++r) {
      const int i = r + 8 * h2;
      Yws[d * (BATCH * LENGTH) + half * LENGTH + c * T + i] = acc[r];
    }
  }
}

// ---------------------------------------------------------------------------
// K4: transpose Yws (D x B*L) -> out (B*L x D), 32x32 LDS tiles; the output
// side streams straight from LDS with GLOBAL_STORE_ASYNC_FROM_LDS_B32.
// ---------------------------------------------------------------------------
__global__ __launch_bounds__(256) void s4d_transpose(
    const float* __restrict__ Yws, float* __restrict__ out) {
  __shared__ float tile[32][33];
  const int tx = threadIdx.x & 31, ty = threadIdx.x >> 5;  // ty 0..7
  const int bt0 = blockIdx.x * 32;                         // time-major index
  const int d0  = blockIdx.y * 32;                         // channel index
#pragma unroll
  for (int k = 0; k < 4; ++k) {
    const int row = ty + k * 8;
    tile[row][tx] = Yws[(d0 + row) * (BATCH * LENGTH) + bt0 + tx];
  }
  __syncthreads();
#pragma unroll
  for (int k = 0; k < 4; ++k) {
    const int row = ty + k * 8;
    async_l2g_b32(&out[(bt0 + row) * D_MODEL + d0 + tx], &tile[tx][row]);
  }
  wait_async0();   // (S_ENDPGM would also wait-idle; explicit for clarity)
}

// ---------------------------------------------------------------------------
extern "C" void kernel_launch(void* const* d_in, const int* in_sizes, int n_in,
                              void* d_out, int out_size, void* d_ws, size_t ws_size,
                              hipStream_t stream) {
  const float* x      = (const float*)d_in[0];   // (2, 2048, 512)
  const float* A_log  = (const float*)d_in[1];   // (64)
  const float* Bmat   = (const float*)d_in[2];   // (512, 64)
  const float* Cmat   = (const float*)d_in[3];   // (512, 64)
  const float* log_dt = (const float*)d_in[4];   // (512)
  float* out = (float*)d_out;                    // (2, 2048, 512)

  float* ws   = (float*)d_ws;
  float* Abar = ws;                                   // 32768
  float* Bbar = Abar + D_MODEL * D_STATE;             // 32768
  float* Mws  = Bbar + D_MODEL * D_STATE;             // 512*16*80
  float* H0   = Mws + D_MODEL * T * KDIM;             // 256*512*64
  float* Yws  = H0 + (size_t)COLS * D_MODEL * D_STATE;// 512*4096

  s4d_params   <<<D_MODEL, 32, 0, stream>>>(A_log, Bmat, Cmat, log_dt, Abar, Bbar, Mws);
  s4d_scan_h0  <<<BATCH * (D_MODEL / 4), 256, 0, stream>>>(x, Abar, Bbar, H0);
  s4d_wmma_ymat<<<D_MODEL, 256, 0, stream>>>(x, Mws, H0, Yws);
  s4d_transpose<<<dim3((BATCH * LENGTH) / 32, D_MODEL / 32), 256, 0, stream>>>(Yws, out);
}